// KoopmanLayer_58454504898659
// MI455X (gfx1250) — compile-verified
//
#include <hip/hip_runtime.h>
#include <hip/hip_bf16.h>

typedef __attribute__((ext_vector_type(2))) float v2f;
typedef __attribute__((ext_vector_type(8))) float v8f;

#define D 64
#define DD (D * D)

// ---------------------------------------------------------------------------
// Phase 1: mats[t] = expm(kernel * time[t])   (one 256-thread block per t)
// Scaling-and-squaring + 12-term Taylor (Horner).  All work in LDS.
// ---------------------------------------------------------------------------

// C = A * B for 64x64 in LDS. 256 threads: thread computes row r = tid>>2,
// 16 contiguous cols starting at (tid&3)*16.
__device__ inline void mm64(float* __restrict__ Cm,
                            const float* __restrict__ Am,
                            const float* __restrict__ Bm, int tid) {
  const int r  = tid >> 2;
  const int c0 = (tid & 3) << 4;
  float4 acc0 = {0.f,0.f,0.f,0.f}, acc1 = acc0, acc2 = acc0, acc3 = acc0;
  for (int k = 0; k < D; ++k) {
    const float a = Am[r * D + k];
    const float4 b0 = *(const float4*)&Bm[k * D + c0 + 0];
    const float4 b1 = *(const float4*)&Bm[k * D + c0 + 4];
    const float4 b2 = *(const float4*)&Bm[k * D + c0 + 8];
    const float4 b3 = *(const float4*)&Bm[k * D + c0 + 12];
    acc0.x = fmaf(a, b0.x, acc0.x); acc0.y = fmaf(a, b0.y, acc0.y);
    acc0.z = fmaf(a, b0.z, acc0.z); acc0.w = fmaf(a, b0.w, acc0.w);
    acc1.x = fmaf(a, b1.x, acc1.x); acc1.y = fmaf(a, b1.y, acc1.y);
    acc1.z = fmaf(a, b1.z, acc1.z); acc1.w = fmaf(a, b1.w, acc1.w);
    acc2.x = fmaf(a, b2.x, acc2.x); acc2.y = fmaf(a, b2.y, acc2.y);
    acc2.z = fmaf(a, b2.z, acc2.z); acc2.w = fmaf(a, b2.w, acc2.w);
    acc3.x = fmaf(a, b3.x, acc3.x); acc3.y = fmaf(a, b3.y, acc3.y);
    acc3.z = fmaf(a, b3.z, acc3.z); acc3.w = fmaf(a, b3.w, acc3.w);
  }
  *(float4*)&Cm[r * D + c0 + 0]  = acc0;
  *(float4*)&Cm[r * D + c0 + 4]  = acc1;
  *(float4*)&Cm[r * D + c0 + 8]  = acc2;
  *(float4*)&Cm[r * D + c0 + 12] = acc3;
}

__global__ __launch_bounds__(256) void koopman_expm_kernel(
    const float* __restrict__ Kmat,   // (64,64)
    const float* __restrict__ timev,  // (T,)
    float* __restrict__ mats)         // (T,64,64) workspace
{
  __shared__ float Xs[DD];
  __shared__ float Ps[DD];
  __shared__ float Cs[DD];
  __shared__ float rowsum[D];
  __shared__ float s_scale;
  __shared__ int   s_sq;

  const int t   = blockIdx.x;
  const int tid = threadIdx.x;
  const float tv = timev[t];

  for (int i = tid; i < DD; i += 256) Xs[i] = Kmat[i];
  __syncthreads();

  // inf-norm of K*t  (max abs row sum)
  if (tid < D) {
    float s = 0.f;
    for (int k = 0; k < D; ++k) s += fabsf(Xs[tid * D + k]);
    rowsum[tid] = s * fabsf(tv);
  }
  __syncthreads();
  if (tid == 0) {
    float nrm = 0.f;
    for (int r = 0; r < D; ++r) nrm = fmaxf(nrm, rowsum[r]);
    int sq = 0;
    while (nrm > 0.5f && sq < 30) { nrm *= 0.5f; ++sq; }
    s_sq = sq;
    s_scale = tv / (float)(1 << sq);
  }
  __syncthreads();
  const float sc = s_scale;
  const int   sq = s_sq;

  // X = K * (t / 2^sq);  P = I
  for (int i = tid; i < DD; i += 256) {
    Xs[i] *= sc;
    Ps[i] = ((i / D) == (i % D)) ? 1.f : 0.f;
  }
  __syncthreads();

  // Horner Taylor: for j = 12..1 : P = I + (X*P)/j   ->  P = exp(X)
  for (int j = 12; j >= 1; --j) {
    mm64(Cs, Xs, Ps, tid);
    __syncthreads();
    const float inv = 1.0f / (float)j;
    for (int i = tid; i < DD; i += 256)
      Ps[i] = Cs[i] * inv + (((i / D) == (i % D)) ? 1.f : 0.f);
    __syncthreads();
  }

  // squaring: P = P^2, sq times
  for (int q = 0; q < sq; ++q) {
    mm64(Cs, Ps, Ps, tid);
    __syncthreads();
    for (int i = tid; i < DD; i += 256) Ps[i] = Cs[i];
    __syncthreads();
  }

  float* dst = mats + (size_t)t * DD;
  for (int i = tid; i < DD; i += 256) dst[i] = Ps[i];
}

// ---------------------------------------------------------------------------
// Phase 2: out[b,t,i] = sum_j mats[t,i,j] * x0[b,j]
// Grid (B/128, T). 256 threads = 8 waves. Block stages mats[t] in LDS.
// Wave w computes rows b0 = bx*128 + 16*w .. +15, all 64 output cols, via
// v_wmma_f32_16x16x4_f32:  A = x0 tile (16x4), B[k][n] = mats[t][n][k].
// ---------------------------------------------------------------------------
__global__ __launch_bounds__(256) void koopman_apply_kernel(
    const float* __restrict__ inputs, // (B,T,64) -- only [:,0,:] used
    const float* __restrict__ mats,   // (T,64,64)
    float* __restrict__ out,          // (B,T,64)
    int T)
{
  __shared__ float Msh[DD];

  const int t   = blockIdx.y;
  const int tid = threadIdx.x;
  const int TD  = T * D;

  const float* Mt = mats + (size_t)t * DD;
  for (int i = tid; i < DD; i += 256) Msh[i] = Mt[i];
  __syncthreads();

  const int wave  = tid >> 5;
  const int lane  = tid & 31;
  const int lhalf = lane >> 4;   // 0: lanes 0-15, 1: lanes 16-31
  const int l16   = lane & 15;
  const int b0    = blockIdx.x * 128 + wave * 16;

  // A fragment addressing (32-bit A 16x4): lane half selects K pair {0,1}/{2,3}
  const float* arow = inputs + (size_t)(b0 + l16) * TD;  // x0 row (t=0 slice)

  // Preload all 16 A fragments (K = 0..63 in steps of 4): 2 floats each, aligned.
  v2f afrag[16];
#pragma unroll
  for (int kk = 0; kk < 16; ++kk)
    afrag[kk] = *(const v2f*)&arow[kk * 4 + 2 * lhalf];

  v8f acc0 = {}, acc1 = {}, acc2 = {}, acc3 = {};

#pragma unroll
  for (int kk = 0; kk < 16; ++kk) {
    const int k0 = kk * 4;
    // B fragment (32-bit B 4x16): b = { B[k0+2*lhalf][n], B[k0+2*lhalf+1][n] }
    // with B[k][n] = Msh[n*64 + k]  (mats[t][i=n][j=k]); contiguous pair in LDS.
    const v2f bf0 = *(const v2f*)&Msh[(0  + l16) * D + k0 + 2 * lhalf];
    const v2f bf1 = *(const v2f*)&Msh[(16 + l16) * D + k0 + 2 * lhalf];
    const v2f bf2 = *(const v2f*)&Msh[(32 + l16) * D + k0 + 2 * lhalf];
    const v2f bf3 = *(const v2f*)&Msh[(48 + l16) * D + k0 + 2 * lhalf];
    const v2f a = afrag[kk];
    acc0 = __builtin_amdgcn_wmma_f32_16x16x4_f32(false, a, false, bf0,
                                                 (short)0, acc0, false, false);
    acc1 = __builtin_amdgcn_wmma_f32_16x16x4_f32(false, a, false, bf1,
                                                 (short)0, acc1, false, false);
    acc2 = __builtin_amdgcn_wmma_f32_16x16x4_f32(false, a, false, bf2,
                                                 (short)0, acc2, false, false);
    acc3 = __builtin_amdgcn_wmma_f32_16x16x4_f32(false, a, false, bf3,
                                                 (short)0, acc3, false, false);
  }

  // C/D layout: VGPR r, lanes 0-15 -> (m=r, n=l16); lanes 16-31 -> (m=r+8, n=l16).
  // Non-temporal: 256MB write-once stream must not evict mats/x0 from L2.
  const size_t obase = (size_t)t * D + l16;
#pragma unroll
  for (int r = 0; r < 8; ++r) {
    const size_t row = (size_t)(b0 + r + 8 * lhalf) * TD + obase;
    __builtin_nontemporal_store(acc0[r], &out[row + 0]);
    __builtin_nontemporal_store(acc1[r], &out[row + 16]);
    __builtin_nontemporal_store(acc2[r], &out[row + 32]);
    __builtin_nontemporal_store(acc3[r], &out[row + 48]);
  }
}

// ---------------------------------------------------------------------------
extern "C" void kernel_launch(void* const* d_in, const int* in_sizes, int n_in,
                              void* d_out, int out_size, void* d_ws, size_t ws_size,
                              hipStream_t stream) {
  const float* inputs = (const float*)d_in[0];  // (B,T,64)
  const float* timev  = (const float*)d_in[1];  // (T,)
  const float* Kmat   = (const float*)d_in[2];  // (64,64)
  float* out  = (float*)d_out;
  float* mats = (float*)d_ws;                   // (T,64,64) scratch: T*16KB

  const int T = in_sizes[1];
  const int B = in_sizes[0] / (T * D);

  koopman_expm_kernel<<<T, 256, 0, stream>>>(Kmat, timev, mats);
  koopman_apply_kernel<<<dim3(B / 128, T), 256, 0, stream>>>(inputs, mats, out, T);
}